// PhysicsInformedNN_18176301597010
// MI455X (gfx1250) — compile-verified
//
#include <hip/hip_runtime.h>

// CDNA5 / gfx1250: wave32, WMMA f32 16x16x4 (the f32 HPC matrix op).
typedef __attribute__((ext_vector_type(2))) float v2f;   // A/B operand: 16x4 f32 = 2 VGPR/lane
typedef __attribute__((ext_vector_type(8))) float v8f;   // C/D operand: 16x16 f32 = 8 VGPR/lane

#define NHID 7
#define WID  20          // real layer width
#define WPAD 32          // padded width = 2 WMMA col tiles
#define NJET 13          // value, x,y,t, xx,xy,yy,xt,yt, xxx,xxy,xyy,yyy
#define WAVES 4
#define PTS_PER_WAVE 16
#define PTS_PER_BLOCK (WAVES*PTS_PER_WAVE)

// ---- LDS layout (float offsets) ----
#define OFF_WT   0                       // transposed hidden weights [7][32][20] : Wt[l][n][k]
#define SZ_WT    (NHID*WPAD*WID)         // 4480
#define OFF_BH   (OFF_WT+SZ_WT)          // hidden bias padded [7][32]
#define SZ_BH    (NHID*WPAD)             // 224
#define OFF_WIN  (OFF_BH+SZ_BH)          // W_in [3][20]
#define OFF_BIN  (OFF_WIN+60)            // b_in [20]
#define OFF_WO   (OFF_BIN+20)            // W_out [20][2]
#define OFF_BO   (OFF_WO+40)             // b_out [2]
#define OFF_MISC (OFF_BO+2)              // lb0..2, inv0..2, lambda1, lambda2
#define OFF_S    ((OFF_MISC+8+15)&~15)   // per-wave jet state, 16-float aligned
#define S_SIZE   (NJET*PTS_PER_WAVE*WID) // 4160 floats per buffer
#define LDS_FLOATS (OFF_S + WAVES*2*S_SIZE)

// Propagate a 3rd-order (x,y,t) jet through tanh via Faa di Bruno.
// z[13] = pre-activation jet, a[13] = activation jet.
__device__ __forceinline__ void tanh_jet(const float* z, float* a) {
    float tv = tanhf(z[0]);
    float s  = 1.f - tv*tv;                 // sigma'
    float d2 = -2.f*tv*s;                   // sigma''
    float d3 = -2.f*s*s + 4.f*tv*tv*s;      // sigma'''
    a[0] = tv;
    a[1] = s*z[1];  a[2] = s*z[2];  a[3] = s*z[3];
    a[4] = d2*z[1]*z[1] + s*z[4];
    a[5] = d2*z[1]*z[2] + s*z[5];
    a[6] = d2*z[2]*z[2] + s*z[6];
    a[7] = d2*z[1]*z[3] + s*z[7];
    a[8] = d2*z[2]*z[3] + s*z[8];
    a[9]  = d3*z[1]*z[1]*z[1] + 3.f*d2*z[4]*z[1] + s*z[9];
    a[10] = d3*z[1]*z[1]*z[2] + d2*(z[4]*z[2] + 2.f*z[5]*z[1]) + s*z[10];
    a[11] = d3*z[1]*z[2]*z[2] + d2*(z[6]*z[1] + 2.f*z[5]*z[2]) + s*z[11];
    a[12] = d3*z[2]*z[2]*z[2] + 3.f*d2*z[6]*z[2] + s*z[12];
}

__global__ __launch_bounds__(PTS_PER_BLOCK*2) void pinn_ns_jet_wmma_kernel(
    const float* __restrict__ x, const float* __restrict__ y, const float* __restrict__ t,
    const float* __restrict__ W_in, const float* __restrict__ b_in,
    const float* __restrict__ W_hid, const float* __restrict__ b_hid,
    const float* __restrict__ W_out, const float* __restrict__ b_out,
    const float* __restrict__ lb, const float* __restrict__ ub,
    const float* __restrict__ lam1, const float* __restrict__ lam2,
    float* __restrict__ out, int N)
{
    extern __shared__ float lds[];
    const int tid   = threadIdx.x;
    const int lane  = tid & 31;
    const int wave  = tid >> 5;
    const int lmod  = lane & 15;   // col (B/C/D) or row (A) within tile
    const int lhalf = lane >> 4;   // selects K/M half per WMMA layout

    // ---- stage weights into LDS (hidden weights transposed: Wt[l][n][k], zero-padded n>=20) ----
    for (int idx = tid; idx < SZ_WT; idx += blockDim.x) {
        int l = idx / (WPAD*WID); int r = idx % (WPAD*WID);
        int n = r / WID; int k = r % WID;
        lds[OFF_WT + idx] = (n < WID) ? W_hid[(l*WID + k)*WID + n] : 0.f;
    }
    for (int idx = tid; idx < SZ_BH; idx += blockDim.x) {
        int l = idx >> 5; int n = idx & 31;
        lds[OFF_BH + idx] = (n < WID) ? b_hid[l*WID + n] : 0.f;
    }
    if (tid < 60) lds[OFF_WIN + tid] = W_in[tid];
    if (tid < 20) lds[OFF_BIN + tid] = b_in[tid];
    if (tid < 40) lds[OFF_WO  + tid] = W_out[tid];
    if (tid < 2)  lds[OFF_BO  + tid] = b_out[tid];
    if (tid < 3) { lds[OFF_MISC + tid] = lb[tid];
                   lds[OFF_MISC + 3 + tid] = 1.f/(ub[tid]-lb[tid]); }
    if (tid == 0){ lds[OFF_MISC + 6] = lam1[0]; lds[OFF_MISC + 7] = lam2[0]; }
    __syncthreads();

    // per-wave double-buffered jet state S[jet][point][neuron] (stride WID keeps v2f loads 8B aligned)
    float* S  = lds + OFF_S + wave*(2*S_SIZE);
    float* S2 = S + S_SIZE;
    const int gbase = blockIdx.x*PTS_PER_BLOCK + wave*PTS_PER_WAVE;

    const float lb0=lds[OFF_MISC+0], lb1=lds[OFF_MISC+1], lb2=lds[OFF_MISC+2];
    const float i0 =lds[OFF_MISC+3], i1 =lds[OFF_MISC+4], i2 =lds[OFF_MISC+5];

    // ---- input layer (K=3, too small for WMMA): direct jet of z = h@W_in + b_in ----
    for (int idx = lane; idx < PTS_PER_WAVE*WID; idx += 32) {
        int pt = idx & 15, n = idx >> 4;
        int g = gbase + pt;
        float xn=(x[g]-lb0)*i0, yn=(y[g]-lb1)*i1, tn=(t[g]-lb2)*i2;
        float w0=lds[OFF_WIN+n], w1=lds[OFF_WIN+WID+n], w2=lds[OFF_WIN+2*WID+n];
        float z[NJET] = {0.f}, a[NJET];
        z[0]=xn*w0 + yn*w1 + tn*w2 + lds[OFF_BIN+n];
        z[1]=i0*w0; z[2]=i1*w1; z[3]=i2*w2;   // higher-order jets of the affine input are 0
        tanh_jet(z, a);
        #pragma unroll
        for (int j=0;j<NJET;++j) S[(j*16+pt)*WID + n] = a[j];
    }

    // ---- 7 hidden layers: batched 208x20 @ 20x20 via WMMA f32 16x16x4 ----
    #pragma unroll 1
    for (int l = 0; l < NHID; ++l) {
        const float* Wt = lds + OFF_WT + l*(WPAD*WID);
        const float* bh = lds + OFF_BH + l*WPAD;
        #pragma unroll
        for (int c = 0; c < 2; ++c) {                 // two 16-wide output-neuron tiles
            int n = 16*c + lmod;
            v2f Bt[5];                                // B tiles: (k,n) lane=n+16*(k/2), vgpr=k&1
            #pragma unroll
            for (int s = 0; s < 5; ++s)
                Bt[s] = *(const v2f*)&Wt[n*WID + 4*s + 2*lhalf];
            float bval = bh[n];
            v8f acc[NJET];
            #pragma unroll
            for (int j = 0; j < NJET; ++j) {          // 13 row tiles = 13 jets x 16 points
                float cinit = (j==0) ? bval : 0.f;    // bias only on the value jet
                #pragma unroll
                for (int v = 0; v < 8; ++v) acc[j][v] = cinit;
                #pragma unroll
                for (int s = 0; s < 5; ++s) {
                    // A tile: (m,k) lane = m + 16*(k/2), vgpr = k&1  -> ds_load_b64
                    v2f A = *(const v2f*)&S[(j*16 + lmod)*WID + 4*s + 2*lhalf];
                    acc[j] = __builtin_amdgcn_wmma_f32_16x16x4_f32(
                        false, A, false, Bt[s], (short)0, acc[j], false, false);
                }
            }
            // D layout: lane holds (point = v+8*lhalf, neuron = n) for v=0..7, all 13 jets in-lane.
            bool active = (n < WID);
            #pragma unroll
            for (int v = 0; v < 8; ++v) {
                int pt = v + 8*lhalf;
                float zj[NJET], aj[NJET];
                #pragma unroll
                for (int j = 0; j < NJET; ++j) zj[j] = acc[j][v];
                tanh_jet(zj, aj);
                if (active) {
                    #pragma unroll
                    for (int j = 0; j < NJET; ++j) S2[(j*16+pt)*WID + n] = aj[j];
                }
            }
        }
        float* tp = S; S = S2; S2 = tp;               // double buffer swap (same-wave LDS is in-order)
    }

    // ---- output layer (N=2) + NS residuals, lanes 0..15 handle one point each ----
    if (lhalf == 0) {
        int pt = lmod;
        int g  = gbase + pt;
        float psi[NJET];
        #pragma unroll
        for (int j = 0; j < NJET; ++j) {
            float acc = (j==0) ? lds[OFF_BO+0] : 0.f;
            #pragma unroll
            for (int k = 0; k < WID; ++k) acc += S[(j*16+pt)*WID + k]*lds[OFF_WO + 2*k + 0];
            psi[j] = acc;
        }
        float pj[3];
        #pragma unroll
        for (int j = 0; j < 3; ++j) {
            float acc = (j==0) ? lds[OFF_BO+1] : 0.f;
            #pragma unroll
            for (int k = 0; k < WID; ++k) acc += S[(j*16+pt)*WID + k]*lds[OFF_WO + 2*k + 1];
            pj[j] = acc;
        }
        float L1 = lds[OFF_MISC+6], L2 = lds[OFF_MISC+7];
        float u   =  psi[2],  vv  = -psi[1];
        float u_x =  psi[5],  u_y =  psi[6],  u_t =  psi[8];
        float v_x = -psi[4],  v_y = -psi[5],  v_t = -psi[7];
        float u_xx =  psi[10], u_yy =  psi[12];
        float v_xx = -psi[9],  v_yy = -psi[11];
        float f_u = u_t + L1*(u*u_x + vv*u_y) + pj[1] - L2*(u_xx + u_yy);
        float f_v = v_t + L1*(u*v_x + vv*v_y) + pj[2] - L2*(v_xx + v_yy);
        out[g]       = u;
        out[N   + g] = vv;
        out[2*N + g] = pj[0];
        out[3*N + g] = f_u;
        out[4*N + g] = f_v;
    }
}

extern "C" void kernel_launch(void* const* d_in, const int* in_sizes, int n_in,
                              void* d_out, int out_size, void* d_ws, size_t ws_size,
                              hipStream_t stream) {
    const float* x     = (const float*)d_in[0];
    const float* y     = (const float*)d_in[1];
    const float* t     = (const float*)d_in[2];
    const float* W_in  = (const float*)d_in[3];
    const float* b_in  = (const float*)d_in[4];
    const float* W_hid = (const float*)d_in[5];
    const float* b_hid = (const float*)d_in[6];
    const float* W_out = (const float*)d_in[7];
    const float* b_out = (const float*)d_in[8];
    const float* lb    = (const float*)d_in[9];
    const float* ub    = (const float*)d_in[10];
    const float* lam1  = (const float*)d_in[11];
    const float* lam2  = (const float*)d_in[12];
    float* out = (float*)d_out;

    int N = in_sizes[0];                       // 262144, divisible by 64
    int blocks = N / PTS_PER_BLOCK;
    size_t shmem = (size_t)LDS_FLOATS * sizeof(float);   // ~152 KB of the 320 KB WGP LDS
    pinn_ns_jet_wmma_kernel<<<blocks, PTS_PER_BLOCK*2, shmem, stream>>>(
        x, y, t, W_in, b_in, W_hid, b_hid, W_out, b_out, lb, ub, lam1, lam2, out, N);
}